// QIntSoftermax_21234318311674
// MI455X (gfx1250) — compile-verified
//
#include <hip/hip_runtime.h>
#include <stdint.h>

// ---------------------------------------------------------------------------
// Quantized "Softermax" over rows of 1024 fp32 values (8,12,1024,1024).
//   x_q   = rndne(min(x*4, 256))/4                    (Q6.2)
//   m_i   = cummax_j<=i ceil(x_q_j),  M = m_last
//   e_i   = 2^(x_q_i - m_i)
//   d     = sum_i e_i * 2^(m_i - M)
//   d_q   = rndne(min(d*64, 65536))/64                (Q10.6)
//   r     = rndne(min((1/d_q)*128, 256))/128          (Q1.7)
//   p_q   = rndne(min(e_i*32768, 65536))/32768        (Q1.15)
//   out_i = rndne(min(p_q * 2^(m_i - M) * r *32768, 65536))/32768
//
// Memory-bound: 805 MB traffic -> ~35 us floor @ 23.3 TB/s.
// One wave32 per row. Rows staged in via CDNA5 async global->LDS DMA
// (ASYNCcnt, th:TH_LOAD_NT since data is read-once and exceeds the 192 MB
// L2), results staged out via async LDS->global DMA so every store
// instruction writes 512 contiguous bytes (4 full HBM lines), th:TH_STORE_NT.
// ---------------------------------------------------------------------------

#define WAVES_PER_BLOCK 8
#define ROW_LEN         1024
#define PER_LANE        32                      // 1024 / 32 lanes
// pad 4 dwords (16 B) per 32 floats: keeps async b128 LDS transfers
// 16B-aligned; per-lane sequential reads (stride 36 dw) have at worst a
// 2-way bank conflict.
#define LDS_ROW_DW      (ROW_LEN + (ROW_LEN / 32) * 4)   // 1152 dwords
#define LDS_ROW_BYTES   (LDS_ROW_DW * 4)                 // 4608 bytes

__global__ __launch_bounds__(256, 1)
void softermax_kernel(const float* __restrict__ x,
                      float* __restrict__ out,
                      int nrows)
{
    __shared__ __align__(16) float tile[WAVES_PER_BLOCK * LDS_ROW_DW];

    const int lane = threadIdx.x & 31;
    const int wave = threadIdx.x >> 5;
    const int row  = blockIdx.x * WAVES_PER_BLOCK + wave;
    if (row >= nrows) return;                  // no barriers used -> safe

    const float* rowp = x   + (size_t)row * ROW_LEN;
    float*       outp = out + (size_t)row * ROW_LEN;

    // Workgroup-relative LDS byte offset of this wave's row buffer.
    // Low 32 bits of a flat shared pointer == LDS offset (aperture encoding).
    const uint32_t lds_row =
        (uint32_t)(uintptr_t)tile + (uint32_t)(wave * LDS_ROW_BYTES);

    // ---- Stage row into LDS via async DMA (ASYNCcnt), non-temporal -------
    // Iteration i moves 512 coalesced bytes: lane L takes global bytes
    // [i*512 + L*16).  Element g = i*128 + L*4 lands at padded LDS dword
    // g + 4*(g>>5)  ->  byte  i*576 + L*16 + (L>>3)*16   (16B aligned).
    #pragma unroll
    for (int i = 0; i < 8; ++i) {
        uint32_t     ldsa = lds_row + (uint32_t)(i * 576 + lane * 16 + (lane >> 3) * 16);
        const float* ga   = rowp + (i * 128 + lane * 4);
        asm volatile("global_load_async_to_lds_b128 %0, %1, off th:TH_LOAD_NT"
                     :: "v"(ldsa), "v"(ga) : "memory");
    }
    asm volatile("s_wait_asynccnt 0" ::: "memory");

    // ---- Pass 1: quantize + per-lane sequential cummax of ceil(x_q) ------
    // Lane L owns contiguous elements [L*32, L*32+32): padded LDS dword
    // L*36 + k  ->  byte L*144 + 4k (16B-aligned b128 reads).
    float* myrow = tile + wave * LDS_ROW_DW + lane * 36;
    const float4* rp = (const float4*)myrow;

    float xq[PER_LANE];   // quantized inputs, later overwritten with e_i
    float mp[PER_LANE];   // running max, later (m_i - M)
    float run = -__builtin_inff();

    #pragma unroll
    for (int j = 0; j < 8; ++j) {
        float4 v4 = rp[j];
        float  vv[4] = {v4.x, v4.y, v4.z, v4.w};
        #pragma unroll
        for (int t = 0; t < 4; ++t) {
            const int k = j * 4 + t;
            float q = __builtin_rintf(fminf(vv[t] * 4.0f, 256.0f)) * 0.25f; // Q6.2
            xq[k] = q;
            run   = fmaxf(run, __builtin_ceilf(q));
            mp[k] = run;
        }
    }

    // ---- wave32 inclusive max-scan of chunk maxima (5 shuffle steps) -----
    float scan = run;
    #pragma unroll
    for (int d = 1; d < 32; d <<= 1) {
        float o = __shfl_up(scan, d, 32);
        if (lane >= d) scan = fmaxf(scan, o);
    }
    const float M = __shfl(scan, 31, 32);      // row max of ceil(x_q)
    float excl = __shfl_up(scan, 1, 32);       // exclusive prefix for this lane
    if (lane == 0) excl = -__builtin_inff();

    // ---- Pass 2: e_i = 2^(x-m_i); accumulate d = sum e_i * 2^(m_i - M) ---
    float acc = 0.0f;
    #pragma unroll
    for (int k = 0; k < PER_LANE; ++k) {
        float m  = fmaxf(excl, mp[k]);
        float e  = __builtin_amdgcn_exp2f(xq[k] - m);    // v_exp_f32
        float dm = m - M;                                // <= 0, integer-valued
        xq[k] = e;
        mp[k] = dm;
        acc += __builtin_ldexpf(e, (int)dm);             // exact *2^(m-M)
    }
    #pragma unroll
    for (int d = 16; d >= 1; d >>= 1)
        acc += __shfl_xor(acc, d, 32);

    const float dq    = __builtin_rintf(fminf(acc * 64.0f, 65536.0f)) * 0.015625f; // Q10.6
    const float rcp   = __builtin_amdgcn_rcpf(dq);
    const float recip = __builtin_rintf(fminf(rcp * 128.0f, 256.0f)) * 0.0078125f; // Q1.7

    // ---- Pass 3: quantize numerator, rescale, write results into the
    //      (now dead) LDS tile in the same padded layout -------------------
    float4* wp = (float4*)myrow;
    #pragma unroll
    for (int j = 0; j < 8; ++j) {
        float4 o4;
        float* po = (float*)&o4;
        #pragma unroll
        for (int t = 0; t < 4; ++t) {
            const int k = j * 4 + t;
            float pq = __builtin_rintf(fminf(xq[k] * 32768.0f, 65536.0f))
                       * (1.0f / 32768.0f);                          // Q1.15
            float o  = __builtin_ldexpf(pq * recip, (int)mp[k]);
            po[t] = __builtin_rintf(fminf(o * 32768.0f, 65536.0f))
                    * (1.0f / 32768.0f);                             // Q1.15
        }
        wp[j] = o4;                                                   // ds_store_b128
    }

    // LDS writes must land before the DMA engine reads them.
    asm volatile("s_wait_dscnt 0" ::: "memory");

    // ---- Stage results out via async LDS->global DMA: each instruction
    //      writes 512 contiguous bytes (4 full 128B lines), non-temporal.
    //      s_endpgm's implicit wait-idle covers the outstanding ASYNCcnt. --
    #pragma unroll
    for (int i = 0; i < 8; ++i) {
        uint32_t ldsa = lds_row + (uint32_t)(i * 576 + lane * 16 + (lane >> 3) * 16);
        float*   ga   = outp + (i * 128 + lane * 4);
        asm volatile("global_store_async_from_lds_b128 %0, %1, off th:TH_STORE_NT"
                     :: "v"(ga), "v"(ldsa) : "memory");
    }
}

extern "C" void kernel_launch(void* const* d_in, const int* in_sizes, int n_in,
                              void* d_out, int out_size, void* d_ws, size_t ws_size,
                              hipStream_t stream)
{
    (void)n_in; (void)out_size; (void)d_ws; (void)ws_size;
    const float* x   = (const float*)d_in[0];   // (8,12,1024,1024) fp32
    // d_in[1] = scale: unused by the reference.
    float*       out = (float*)d_out;

    const int nrows  = in_sizes[0] / ROW_LEN;   // 98304
    const int blocks = (nrows + WAVES_PER_BLOCK - 1) / WAVES_PER_BLOCK;

    softermax_kernel<<<blocks, WAVES_PER_BLOCK * 32, 0, stream>>>(x, out, nrows);
}